// CenternetPostprocess_60567628808387
// MI455X (gfx1250) — compile-verified
//
#include <hip/hip_runtime.h>
#include <stdint.h>

// CenterNet postprocess for MI455X (gfx1250).
// 2 streaming NMS passes (linear-bin histogram -> collect) using async
// global->LDS staging, then per-batch bitonic top-100 + gather.

typedef unsigned int u32;
typedef unsigned long long u64;

#define B_      32
#define NC      80          // classes
#define NCH     84          // channels (80 heatmap + 4 coors)
#define FH_     128
#define FW_     128
#define SP_     (FH_ * FW_) // 16384
#define K_      100
#define CAP     8192        // candidate buffer per batch
#define NBIN    1024
#define CG      8           // classes per block
#define RB      8           // rows per block
#define NBANDS  (FH_ / RB)  // 16
#define NCGRP   (NC / CG)   // 10
#define TPB     256

__device__ __forceinline__ void async_load_b128(u32 lds_byte_off,
                                                const void* sbase,
                                                u32 voff_bytes) {
  // GVS mode: mem = SGPR64 base + VGPR32 offset; dst = LDS byte address.
  asm volatile("global_load_async_to_lds_b128 %0, %1, %2"
               :: "v"(lds_byte_off), "v"(voff_bytes), "s"(sbase)
               : "memory");
}

__device__ __forceinline__ void wait_async_zero() {
  asm volatile("s_wait_asynccnt 0" ::: "memory");
}

__device__ __forceinline__ u32 qbin_of(float s) {
  // Monotonic linear quantization of s in [0,1) -> [0,1023].
  u32 q = (u32)(s * (float)NBIN);
  return q > (NBIN - 1) ? (NBIN - 1) : q;
}

// MODE 0: histogram of linear bins
// MODE 1: collect candidates with qbin >= bin1
template <int MODE>
__global__ __launch_bounds__(TPB) void nms_pass(
    const float* __restrict__ Y, u32* __restrict__ hist,
    const u32* __restrict__ meta, u32* __restrict__ candCnt,
    u64* __restrict__ cand) {
  __shared__ float tile[CG * (RB + 2) * FW_]; // 8*10*128 f32 = 40 KB
  __shared__ u32 shist[NBIN];

  const int band = blockIdx.x;
  const int cg   = blockIdx.y;
  const int b    = blockIdx.z;
  const int tid  = threadIdx.x;

  const float* base = Y + ((size_t)b * NCH + (size_t)cg * CG) * SP_;

  if (MODE == 0)
    for (int i = tid; i < NBIN; i += TPB) shist[i] = 0;

  // ---- async stage: 8 ch x 10 rows (with halo, row-clamped) x 128 cols ----
  const int NLOAD = CG * (RB + 2) * FW_ / 4; // 2560 x b128
  for (int i = tid; i < NLOAD; i += TPB) {
    int ch  = i / ((RB + 2) * FW_ / 4);
    int rem = i - ch * ((RB + 2) * FW_ / 4);
    int r   = rem / (FW_ / 4);
    int xq  = rem - r * (FW_ / 4);
    int row = band * RB - 1 + r;
    row = row < 0 ? 0 : (row > FH_ - 1 ? FH_ - 1 : row);
    u32 lds_off = (u32)(uintptr_t)&tile[(ch * (RB + 2) + r) * FW_ + xq * 4];
    u32 voff    = (u32)((ch * SP_ + row * FW_ + xq * 4) * 4);
    async_load_b128(lds_off, base, voff);
  }
  wait_async_zero();
  __syncthreads();

  // ---- halo rows outside the image are -inf (reduce_window SAME init) ----
  if (band == 0) {
    for (int i = tid; i < CG * FW_; i += TPB) {
      int ch = i >> 7, x = i & (FW_ - 1);
      tile[(ch * (RB + 2) + 0) * FW_ + x] = __int_as_float(0xff800000);
    }
  }
  if (band == NBANDS - 1) {
    for (int i = tid; i < CG * FW_; i += TPB) {
      int ch = i >> 7, x = i & (FW_ - 1);
      tile[(ch * (RB + 2) + RB + 1) * FW_ + x] = __int_as_float(0xff800000);
    }
  }
  __syncthreads();

  u32 bin1 = 0;
  if (MODE == 1) bin1 = meta[b * 16 + 0];

  const int NEL = CG * RB * FW_; // 8192 -> 32 per thread
  for (int j = tid; j < NEL; j += TPB) {
    int lc  = j >> 10;
    int rem = j & 1023;
    int lr  = rem >> 7;
    int x   = rem & 127;
    const float* t = &tile[(lc * (RB + 2) + lr + 1) * FW_ + x];
    float v = t[0];
    float m = v;
#pragma unroll
    for (int dr = -1; dr <= 1; ++dr) {
      const float* tr = t + dr * FW_;
      m = fmaxf(m, tr[0]);
      if (x > 0)        m = fmaxf(m, tr[-1]);
      if (x < FW_ - 1)  m = fmaxf(m, tr[1]);
    }
    float s  = (v == m) ? v : 0.0f;
    u32 bits = __float_as_uint(s);
    if (bits == 0u) continue; // zeros can never reach top-100 on real data
    u32 q = qbin_of(s);
    if (MODE == 0) {
      atomicAdd(&shist[q], 1u);
    } else {
      if (q >= bin1) {
        int gy  = band * RB + lr;
        int c   = cg * CG + lc;
        u32 ind = (u32)((gy * FW_ + x) * NC + c); // HWC flatten, matches ref
        u32 pos = atomicAdd(&candCnt[b], 1u);
        if (pos < CAP)
          cand[(size_t)b * CAP + pos] =
              ((u64)bits << 32) | (u64)(0xFFFFFFFFu - ind);
      }
    }
  }

  if (MODE == 0) {
    __syncthreads();
    for (int i = tid; i < NBIN; i += TPB) {
      u32 h = shist[i];
      if (h) atomicAdd(&hist[b * NBIN + i], h);
    }
  }
}

__global__ void thresh_pass(const u32* __restrict__ hist,
                            u32* __restrict__ meta) {
  int b = blockIdx.x;
  if (threadIdx.x != 0) return;
  u32 acc = 0;
  const u32* h = hist + b * NBIN;
  int bin = NBIN - 1;
  for (; bin > 0; --bin) {
    u32 c = h[bin];
    if (acc + c >= (u32)K_) break;
    acc += c;
  }
  meta[b * 16 + 0] = (u32)bin;
}

__global__ __launch_bounds__(TPB) void final_topk(
    const float* __restrict__ Y, const u32* __restrict__ candCnt,
    const u64* __restrict__ cand, float* __restrict__ out) {
  __shared__ u64 keys[CAP]; // 64 KB
  int b = blockIdx.x, tid = threadIdx.x;
  u32 M = candCnt[b];
  if (M > CAP) M = CAP;
  for (int i = tid; i < CAP; i += TPB)
    keys[i] = (i < (int)M) ? cand[(size_t)b * CAP + i] : 0ull;
  __syncthreads();

  // bitonic sort, descending (score desc, then index asc via ~ind)
  for (int k2 = 2; k2 <= CAP; k2 <<= 1) {
    for (int j = k2 >> 1; j > 0; j >>= 1) {
      for (int i = tid; i < CAP; i += TPB) {
        int ixj = i ^ j;
        if (ixj > i) {
          u64 a = keys[i], c = keys[ixj];
          bool up = ((i & k2) == 0);
          if (up ? (a < c) : (a > c)) { keys[i] = c; keys[ixj] = a; }
        }
      }
      __syncthreads();
    }
  }

  for (int i = tid; i < K_; i += TPB) {
    u64 key     = keys[i];
    float score = __uint_as_float((u32)(key >> 32));
    u32 ind     = 0xFFFFFFFFu - (u32)(key & 0xFFFFFFFFull);
    if (ind >= (u32)(NC * SP_)) ind = 0; // padding guard
    int c  = (int)(ind % NC);
    int sp = (int)(ind / NC);
    float xs = (float)(sp % FW_);
    float ys = (float)(sp / FW_);
    const float* cb = Y + ((size_t)b * NCH + NC) * SP_ + sp;
    float g0 = cb[0], g1 = cb[SP_], g2 = cb[2 * SP_], g3 = cb[3 * SP_];
    float* o = out + ((size_t)b * K_ + i) * 8;
    o[0] = (float)c + 1.0f;
    o[1] = score;
    o[2] = (4.0f * xs - g0) * (1.0f / 512.0f);
    o[3] = (4.0f * ys - g1) * (1.0f / 512.0f);
    o[4] = (4.0f * xs + g2) * (1.0f / 512.0f);
    o[5] = (4.0f * ys + g3) * (1.0f / 512.0f);
    o[6] = ys;
    o[7] = xs;
  }
}

extern "C" void kernel_launch(void* const* d_in, const int* in_sizes, int n_in,
                              void* d_out, int out_size, void* d_ws,
                              size_t ws_size, hipStream_t stream) {
  (void)in_sizes; (void)n_in; (void)out_size; (void)ws_size;
  const float* Y = (const float*)d_in[0];
  float* out     = (float*)d_out;

  u32* hist    = (u32*)d_ws;               // B*1024
  u32* meta    = hist + B_ * NBIN;         // B*16
  u32* candCnt = meta + B_ * 16;           // B
  size_t zero_words = (size_t)B_ * NBIN + B_ * 16 + B_;
  u64* cand = (u64*)(((uintptr_t)(candCnt + B_) + 7) & ~(uintptr_t)7);

  hipMemsetAsync(d_ws, 0, zero_words * 4, stream);

  dim3 grid(NBANDS, NCGRP, B_); // 16 x 10 x 32 = 5120 blocks
  nms_pass<0><<<grid, TPB, 0, stream>>>(Y, hist, meta, candCnt, cand);
  thresh_pass<<<B_, 32, 0, stream>>>(hist, meta);
  nms_pass<1><<<grid, TPB, 0, stream>>>(Y, hist, meta, candCnt, cand);
  final_topk<<<B_, TPB, 0, stream>>>(Y, candCnt, cand, out);
}